// TopK_90391881712138
// MI455X (gfx1250) — compile-verified
//
#include <hip/hip_runtime.h>
#include <stdint.h>

// ---------------------------------------------------------------------------
// Top-K(32) per row + ReLU + scatter, rows of 65536 fp32.
// One workgroup (1024 threads = 32 wave32) per row.
// Row is staged once into LDS with the gfx1250 async global->LDS engine,
// then an exact 4x8-bit radix select finds the K-th largest key, then a
// single write pass emits the sparse result with non-temporal b128 stores.
// HBM traffic = 1 read + 1 write (~2 GiB / 23.3 TB/s ~= 92 us floor).
// ---------------------------------------------------------------------------

constexpr int N_COLS   = 65536;
constexpr int TOPK     = 32;
constexpr int NTHREADS = 1024;          // 32 waves (wave32)
constexpr int NWAVES   = NTHREADS / 32;
constexpr int EPT      = N_COLS / NTHREADS;  // 64 elements / thread
constexpr int CHUNKS   = EPT / 4;            // 16 x uint4 / thread

#if defined(__AMDGCN__) && __has_builtin(__builtin_amdgcn_global_load_async_to_lds_b128)
#define USE_ASYNC_LDS 1
#else
#define USE_ASYNC_LDS 0
#endif

#if USE_ASYNC_LDS
typedef __attribute__((ext_vector_type(4))) int v4i;
typedef __attribute__((address_space(1))) v4i* g_v4i_p;   // global (AS1) v4i*
typedef __attribute__((address_space(3))) v4i* l_v4i_p;   // LDS    (AS3) v4i*
#endif

typedef __attribute__((ext_vector_type(4))) float f4;     // for NT stores

// monotonic fp32 -> u32 key: bigger float => bigger unsigned key
__device__ __forceinline__ uint32_t f2key(uint32_t b) {
  return b ^ (uint32_t)(((int32_t)b >> 31) | (int32_t)0x80000000);
}

__device__ __forceinline__ float relu_of(uint32_t bits) {
  float f = __uint_as_float(bits);
  return f > 0.0f ? f : 0.0f;
}

// Block-wide: given sHist[256] (bin = key byte, larger bin = larger key) and
// *sNeeded, find byte b with count(byte>b) < needed <= count(byte>=b).
// Publishes b in *sSel and needed-count(byte>b) in *sNeeded.
__device__ __forceinline__ void select_byte(int t, uint32_t* sHist, uint32_t* sScan,
                                            uint32_t* sSel, uint32_t* sNeeded) {
  const uint32_t needed = *sNeeded;           // all threads read before any write
  if (t < 256) sScan[t] = sHist[t];
  __syncthreads();
  // inclusive suffix-sum over 256 bins (Hillis-Steele, 8 steps)
  for (int off = 1; off < 256; off <<= 1) {
    uint32_t v = 0;
    if (t < 256) { v = sScan[t]; if (t + off < 256) v += sScan[t + off]; }
    __syncthreads();
    if (t < 256) sScan[t] = v;
    __syncthreads();
  }
  if (t < 256) {
    const uint32_t ge = sScan[t];                          // count(byte >= t)
    const uint32_t gt = (t < 255) ? sScan[t + 1] : 0u;     // count(byte >  t)
    if (gt < needed && needed <= ge) { *sSel = (uint32_t)t; *sNeeded = needed - gt; }
  }
  __syncthreads();
}

__device__ __forceinline__ float emit_val(uint32_t bits, uint32_t thrKey,
                                          uint32_t needEq, uint32_t* sTie) {
  const uint32_t k = f2key(bits);
  if (k > thrKey) return relu_of(bits);
  if (k == thrKey) {
    if (atomicAdd(sTie, 1u) < needEq) return relu_of(bits);
  }
  return 0.0f;
}

extern "C" __global__ __launch_bounds__(NTHREADS, 1)
void topk_relu_scatter(const float* __restrict__ x, float* __restrict__ out, int nrows) {
  extern __shared__ __align__(16) uint32_t sRow[];   // N_COLS words = 256 KB

  __shared__ uint32_t sHistW[NWAVES * 257];          // wave-private pass-1 hist (padded stride)
  __shared__ uint32_t sHist[256];
  __shared__ uint32_t sScan[256];
  __shared__ uint32_t sSel;
  __shared__ uint32_t sNeeded;
  __shared__ uint32_t sTie;

  const int t   = threadIdx.x;
  const int row = blockIdx.x;
  if (row >= nrows) return;
  const float* __restrict__ xrow = x   + (size_t)row * N_COLS;
  float*       __restrict__ orow = out + (size_t)row * N_COLS;

  // ---- Stage the whole row into LDS ------------------------------------
#if USE_ASYNC_LDS
  for (int c = 0; c < CHUNKS; ++c) {
    const int off = c * (NTHREADS * 4) + t * 4;
    g_v4i_p g = (g_v4i_p)(uintptr_t)(xrow + off);
    l_v4i_p l = (l_v4i_p)(uint32_t)(uintptr_t)(&sRow[off]);
    __builtin_amdgcn_global_load_async_to_lds_b128(g, l, 0, 0);
  }
#else
  for (int c = 0; c < CHUNKS; ++c) {
    const int off = c * (NTHREADS * 4) + t * 4;
    *(uint4*)(&sRow[off]) = *(const uint4*)(xrow + off);
  }
#endif

  // init histograms while the async engine streams data
  for (int i = t; i < NWAVES * 257; i += NTHREADS) sHistW[i] = 0;
  if (t < 256) sHist[t] = 0;
  if (t == 0) { sNeeded = TOPK; sTie = 0; }

#if USE_ASYNC_LDS
  #if __has_builtin(__builtin_amdgcn_s_wait_asynccnt)
  __builtin_amdgcn_s_wait_asynccnt(0);
  #else
  asm volatile("s_wait_asynccnt 0" ::: "memory");
  #endif
#endif
  __syncthreads();

  // ---- Pass 1: histogram of top key byte (wave-private, contention-safe) --
  uint32_t* myHist = &sHistW[(t >> 5) * 257];
  for (int c = 0; c < CHUNKS; ++c) {
    const int off = c * (NTHREADS * 4) + t * 4;
    const uint4 v = *(const uint4*)(&sRow[off]);
    atomicAdd(&myHist[f2key(v.x) >> 24], 1u);
    atomicAdd(&myHist[f2key(v.y) >> 24], 1u);
    atomicAdd(&myHist[f2key(v.z) >> 24], 1u);
    atomicAdd(&myHist[f2key(v.w) >> 24], 1u);
  }
  __syncthreads();
  if (t < 256) {
    uint32_t s = 0;
    #pragma unroll
    for (int w = 0; w < NWAVES; ++w) s += sHistW[w * 257 + t];
    sHist[t] = s;
  }
  __syncthreads();
  select_byte(t, sHist, sScan, &sSel, &sNeeded);
  uint32_t prefix = sSel;

  // ---- Passes 2..4: refine one key byte at a time ----------------------
  for (int p = 1; p < 4; ++p) {
    if (t < 256) sHist[t] = 0;
    __syncthreads();
    const int shiftP = 32 - 8 * p;   // bits of prefix decided so far
    const int shiftB = 24 - 8 * p;   // byte under test
    for (int c = 0; c < CHUNKS; ++c) {
      const int off = c * (NTHREADS * 4) + t * 4;
      const uint4 v = *(const uint4*)(&sRow[off]);
      uint32_t k;
      k = f2key(v.x); if ((k >> shiftP) == prefix) atomicAdd(&sHist[(k >> shiftB) & 0xFFu], 1u);
      k = f2key(v.y); if ((k >> shiftP) == prefix) atomicAdd(&sHist[(k >> shiftB) & 0xFFu], 1u);
      k = f2key(v.z); if ((k >> shiftP) == prefix) atomicAdd(&sHist[(k >> shiftB) & 0xFFu], 1u);
      k = f2key(v.w); if ((k >> shiftP) == prefix) atomicAdd(&sHist[(k >> shiftB) & 0xFFu], 1u);
    }
    __syncthreads();
    select_byte(t, sHist, sScan, &sSel, &sNeeded);
    prefix = (prefix << 8) | sSel;
  }

  const uint32_t thrKey = prefix;    // exact 32-bit key of the K-th largest
  const uint32_t needEq = sNeeded;   // how many elements == thrKey to keep

  // ---- Write pass: zeros + ReLU'd winners, non-temporal b128 stores ----
  for (int c = 0; c < CHUNKS; ++c) {
    const int off = c * (NTHREADS * 4) + t * 4;
    const uint4 v = *(const uint4*)(&sRow[off]);
    f4 o;
    o.x = emit_val(v.x, thrKey, needEq, &sTie);
    o.y = emit_val(v.y, thrKey, needEq, &sTie);
    o.z = emit_val(v.z, thrKey, needEq, &sTie);
    o.w = emit_val(v.w, thrKey, needEq, &sTie);
    __builtin_nontemporal_store(o, (f4*)(orow + off));   // streaming store, skip L2 retention
  }
}

extern "C" void kernel_launch(void* const* d_in, const int* in_sizes, int n_in,
                              void* d_out, int out_size, void* d_ws, size_t ws_size,
                              hipStream_t stream) {
  const float* x = (const float*)d_in[0];
  float* out = (float*)d_out;
  const int rows = in_sizes[0] / N_COLS;
  const size_t shmem = (size_t)N_COLS * sizeof(uint32_t);   // 256 KB dynamic LDS
  (void)n_in; (void)out_size; (void)d_ws; (void)ws_size;
  (void)hipFuncSetAttribute((const void*)topk_relu_scatter,
                            hipFuncAttributeMaxDynamicSharedMemorySize, (int)shmem);
  topk_relu_scatter<<<dim3(rows), dim3(NTHREADS), shmem, stream>>>(x, out, rows);
}